// AttentiveFP_77360950936127
// MI455X (gfx1250) — compile-verified
//
#include <hip/hip_runtime.h>

#define NN 50000
#define EE 800000
#define IND 64
#define HH 128
#define EDD 16
#define BBG 2048
#define LLAYERS 2
#define NTS 2

typedef __attribute__((ext_vector_type(2))) float v2f;
typedef __attribute__((ext_vector_type(8))) float v8f;

__device__ __forceinline__ float leakyf(float v) { return v > 0.f ? v : 0.01f * v; }
__device__ __forceinline__ float eluf(float v)   { return v > 0.f ? v : expm1f(v); }
__device__ __forceinline__ float sigmf(float v)  { return 1.f / (1.f + expf(-v)); }

enum { ACT_NONE = 0, ACT_LEAKY = 1, ACT_ELU = 2, ACT_RELU = 3 };

__device__ __forceinline__ void atomicMaxF(float* addr, float val) {
  int old = __float_as_int(*addr);
  while (__int_as_float(old) < val) {
    int prev = atomicCAS((int*)addr, old, __float_as_int(val));
    if (prev == old) break;
    old = prev;
  }
}

// ---------------------------------------------------------------------------
// WMMA GEMM: C[M,Out] = act(A[M,K] @ W[Out,K]^T + bias)
// fp32 WMMA 16x16x4. Block = 128 threads = 4 waves; the block stages one
// 64-column weight strip (64 x K floats, K<=128 -> <=33KB) into LDS with a
// padded stride (K+4) so B-fragment ds_load_b64s are bank-conflict-free
// (half-wave lanes hit banks 0,4,...,60). Each wave computes a 16x64 strip
// (4 accumulators): per k-step, 1 global b64 (A frag) + 4 LDS b64 (B frags)
// feed 4 back-to-back WMMAs on independent accumulators.
// Requires Out % 64 == 0, K % 4 == 0 (K <= 128), M % 16 == 0.
// A-frag (ISA 7.12.2, 32-bit A 16x4): lanes0-15 M=lane,K={k0,k0+1};
// lanes16-31 K={k0+2,k0+3}. B-frag mirrors. C/D: v8f, lanes0-15 -> M=r,
// lanes16-31 -> M=r+8, N=lane&15.
// ---------------------------------------------------------------------------
template <int ACT>
__global__ __launch_bounds__(128) void gemm_k(const float* __restrict__ A, int lda,
                                              const float* __restrict__ W, int ldw,
                                              const float* __restrict__ bias,
                                              float* __restrict__ C, int ldc,
                                              int M, int K, int Out) {
  __shared__ float sW[64 * (HH + 4)];  // 64 cols x (K+4) stride, K<=128
  const int tid = threadIdx.x;
  const int col0 = blockIdx.y * 64;
  const int SP = K + 4;  // padded LDS stride (floats)
  // ---- stage the 64 x K weight strip into LDS (coalesced float4) ----
  const int kq = K >> 2;  // float4s per weight row
  for (int idx = tid; idx < 64 * kq; idx += 128) {
    int c = idx / kq;
    int kk = (idx - c * kq) << 2;
    const float4 vv = *(const float4*)(W + (size_t)(col0 + c) * ldw + kk);
    *(float4*)&sW[c * SP + kk] = vv;
  }
  __syncthreads();

  const int lane = tid & 31;
  const int wave = tid >> 5;
  const int row0 = (blockIdx.x * 4 + wave) * 16;
  if (row0 >= M) return;  // after the barrier: safe
  const int lh = lane & 15;
  const int kh = (lane >> 4) << 1;  // 0 or 2
  v8f c0 = {0.f, 0.f, 0.f, 0.f, 0.f, 0.f, 0.f, 0.f};
  v8f c1 = c0, c2 = c0, c3 = c0;
  const float* arow = A + (size_t)(row0 + lh) * lda + kh;
  const float* s0 = &sW[(0 + lh) * SP + kh];
  const float* s1 = &sW[(16 + lh) * SP + kh];
  const float* s2 = &sW[(32 + lh) * SP + kh];
  const float* s3 = &sW[(48 + lh) * SP + kh];
  __builtin_prefetch(arow, 0, 1);
  #pragma unroll 4
  for (int k0 = 0; k0 < K; k0 += 4) {
    v2f av, b0, b1, b2, b3;
    av.x = arow[k0 + 0]; av.y = arow[k0 + 1];
    b0.x = s0[k0 + 0];   b0.y = s0[k0 + 1];
    b1.x = s1[k0 + 0];   b1.y = s1[k0 + 1];
    b2.x = s2[k0 + 0];   b2.y = s2[k0 + 1];
    b3.x = s3[k0 + 0];   b3.y = s3[k0 + 1];
    c0 = __builtin_amdgcn_wmma_f32_16x16x4_f32(false, av, false, b0, (short)0, c0, false, false);
    c1 = __builtin_amdgcn_wmma_f32_16x16x4_f32(false, av, false, b1, (short)0, c1, false, false);
    c2 = __builtin_amdgcn_wmma_f32_16x16x4_f32(false, av, false, b2, (short)0, c2, false, false);
    c3 = __builtin_amdgcn_wmma_f32_16x16x4_f32(false, av, false, b3, (short)0, c3, false, false);
  }
  const int mo = (lane >> 4) << 3;  // +8 rows for upper half-wave
  #pragma unroll
  for (int j = 0; j < 4; ++j) {
    v8f cc = (j == 0) ? c0 : (j == 1) ? c1 : (j == 2) ? c2 : c3;
    const int col = col0 + j * 16 + lh;
    const float bb = bias ? bias[col] : 0.f;
    #pragma unroll
    for (int r = 0; r < 8; ++r) {
      float v = cc[r] + bb;
      if (ACT == ACT_LEAKY) v = leakyf(v);
      else if (ACT == ACT_ELU) v = eluf(v);
      else if (ACT == ACT_RELU) v = fmaxf(v, 0.f);
      C[(size_t)(row0 + mo + r) * ldc + col] = v;
    }
  }
}

// ov[k] = sum_h w[h][k] * a[h]   (w is [128,128] row-major) -- one block of 128
__global__ void matvecT_k(const float* __restrict__ w, const float* __restrict__ a,
                          float* __restrict__ ov) {
  int k = threadIdx.x;
  float s = 0.f;
  for (int h = 0; h < HH; ++h) s += w[h * HH + k] * a[h];
  ov[k] = s;
}

// per-row dual dot against two 128-vectors; wave per row
__global__ __launch_bounds__(128) void dot2_k(const float* __restrict__ X,
                                              const float* __restrict__ v1,
                                              const float* __restrict__ v2,
                                              float* __restrict__ o1,
                                              float* __restrict__ o2, int n) {
  int lane = threadIdx.x & 31;
  int row = blockIdx.x * 4 + (threadIdx.x >> 5);
  if (row >= n) return;
  float s1 = 0.f, s2 = 0.f;
  #pragma unroll
  for (int i = 0; i < 4; ++i) {
    int idx = i * 32 + lane;
    float x = X[(size_t)row * HH + idx];
    s1 += x * v1[idx];
    s2 += x * v2[idx];
  }
  for (int off = 16; off; off >>= 1) {
    s1 += __shfl_xor(s1, off, 32);
    s2 += __shfl_xor(s2, off, 32);
  }
  if (lane == 0) { o1[row] = s1; o2[row] = s2; }
}

__global__ void fill_k(float* __restrict__ p, float v, int n) {
  int i = blockIdx.x * 256 + threadIdx.x;
  if (i < n) p[i] = v;
}

// pass 1: raw attention logit + running segment max (src==nullptr -> identity)
__global__ void edge_alpha_k(const int* __restrict__ src, const int* __restrict__ dst,
                             const float* __restrict__ ssrc, const float* __restrict__ sdst,
                             float* __restrict__ araw, float* __restrict__ nmax, int n) {
  int e = blockIdx.x * 256 + threadIdx.x;
  if (e >= n) return;
  int s = src ? src[e] : e;
  int d = dst[e];
  float a = leakyf(ssrc[s] + sdst[d]);
  araw[e] = a;
  atomicMaxF(&nmax[d], a);
}

// pass 2: exp(logit - segmax) + segment sum
__global__ void edge_exp_k(const int* __restrict__ dst, const float* __restrict__ araw,
                           const float* __restrict__ nmax, float* __restrict__ aexp,
                           float* __restrict__ nsum, int n) {
  int e = blockIdx.x * 256 + threadIdx.x;
  if (e >= n) return;
  int d = dst[e];
  float w = expf(araw[e] - nmax[d]);
  aexp[e] = w;
  atomicAdd(&nsum[d], w);
}

// pass 3: acc[dst] += (aexp/nsum[dst]) * X[src];  wave per edge
__global__ __launch_bounds__(128) void edge_scatter_k(const int* __restrict__ src,
                                                      const int* __restrict__ dst,
                                                      const float* __restrict__ aexp,
                                                      const float* __restrict__ nsum,
                                                      const float* __restrict__ X,
                                                      float* __restrict__ acc, int n) {
  int lane = threadIdx.x & 31;
  int e = blockIdx.x * 4 + (threadIdx.x >> 5);
  if (e >= n) return;
  int s = src ? src[e] : e;
  int d = dst[e];
  float coeff = aexp[e] / nsum[d];
  #pragma unroll
  for (int i = 0; i < 4; ++i) {
    int idx = i * 32 + lane;
    atomicAdd(&acc[(size_t)d * HH + idx], coeff * X[(size_t)s * HH + idx]);
  }
}

// acc[batch[n]] += X[n];  wave per node
__global__ __launch_bounds__(128) void segsum_k(const int* __restrict__ batch,
                                                const float* __restrict__ X,
                                                float* __restrict__ acc, int n) {
  int lane = threadIdx.x & 31;
  int r = blockIdx.x * 4 + (threadIdx.x >> 5);
  if (r >= n) return;
  int g = batch[r];
  #pragma unroll
  for (int i = 0; i < 4; ++i) {
    int idx = i * 32 + lane;
    atomicAdd(&acc[(size_t)g * HH + idx], X[(size_t)r * HH + idx]);
  }
}

__global__ void relu_copy_k(const float* __restrict__ in, float* __restrict__ out, int n) {
  int i = blockIdx.x * 256 + threadIdx.x;
  if (i < n) out[i] = fmaxf(in[i], 0.f);
}

// GATEConv edge pass 1: recompute m_e = leaky(xw1[src] + ea@W1b^T), dot with att_l
__global__ __launch_bounds__(128) void gate_edge1_k(const int* __restrict__ ei,
                                                    const float* __restrict__ g_lin1_w,
                                                    const float* __restrict__ g_att_l,
                                                    const float* __restrict__ xw1,
                                                    const float* __restrict__ edge_attr,
                                                    const float* __restrict__ xr,
                                                    float* __restrict__ araw,
                                                    float* __restrict__ nmax) {
  __shared__ float sW[HH * EDD];
  __shared__ float sA[HH];
  for (int i = threadIdx.x; i < HH * EDD; i += 128)
    sW[i] = g_lin1_w[(i >> 4) * (HH + EDD) + HH + (i & 15)];
  if (threadIdx.x < HH) sA[threadIdx.x] = g_att_l[threadIdx.x];
  __syncthreads();
  int lane = threadIdx.x & 31;
  int e = blockIdx.x * 4 + (threadIdx.x >> 5);
  if (e >= EE) return;
  int s = ei[e], d = ei[EE + e];
  float eav[EDD];
  const float* ea = edge_attr + (size_t)e * EDD;
  #pragma unroll
  for (int j = 0; j < EDD; ++j) eav[j] = ea[j];
  float dot = 0.f;
  #pragma unroll
  for (int i = 0; i < 4; ++i) {
    int hh = i * 32 + lane;
    float mh = xw1[(size_t)s * HH + hh];
    const float* wr = &sW[hh * EDD];
    #pragma unroll
    for (int j = 0; j < EDD; ++j) mh += eav[j] * wr[j];
    mh = leakyf(mh);
    dot += mh * sA[hh];
  }
  for (int off = 16; off; off >>= 1) dot += __shfl_xor(dot, off, 32);
  if (lane == 0) {
    float a = leakyf(dot + xr[d]);
    araw[e] = a;
    atomicMaxF(&nmax[d], a);
  }
}

// GATEConv edge pass 3: acc[dst] += coeff * m_e (m recomputed)
__global__ __launch_bounds__(128) void gate_edge3_k(const int* __restrict__ ei,
                                                    const float* __restrict__ g_lin1_w,
                                                    const float* __restrict__ xw1,
                                                    const float* __restrict__ edge_attr,
                                                    const float* __restrict__ aexp,
                                                    const float* __restrict__ nsum,
                                                    float* __restrict__ acc) {
  __shared__ float sW[HH * EDD];
  for (int i = threadIdx.x; i < HH * EDD; i += 128)
    sW[i] = g_lin1_w[(i >> 4) * (HH + EDD) + HH + (i & 15)];
  __syncthreads();
  int lane = threadIdx.x & 31;
  int e = blockIdx.x * 4 + (threadIdx.x >> 5);
  if (e >= EE) return;
  int s = ei[e], d = ei[EE + e];
  float coeff = aexp[e] / nsum[d];
  float eav[EDD];
  const float* ea = edge_attr + (size_t)e * EDD;
  #pragma unroll
  for (int j = 0; j < EDD; ++j) eav[j] = ea[j];
  #pragma unroll
  for (int i = 0; i < 4; ++i) {
    int hh = i * 32 + lane;
    float mh = xw1[(size_t)s * HH + hh];
    const float* wr = &sW[hh * EDD];
    #pragma unroll
    for (int j = 0; j < EDD; ++j) mh += eav[j] * wr[j];
    mh = leakyf(mh);
    atomicAdd(&acc[(size_t)d * HH + hh], coeff * mh);
  }
}

// GRU gate combine + relu: xo = relu((1-z)*tanh(in+r*hn) + z*hprev)
__global__ void gru_combine_k(const float* __restrict__ gi, const float* __restrict__ gh,
                              const float* __restrict__ hprev, float* __restrict__ xo,
                              int rows) {
  int i = blockIdx.x * 256 + threadIdx.x;
  if (i >= rows * HH) return;
  int r = i >> 7, h = i & 127;
  size_t b3 = (size_t)r * (3 * HH);
  float ir = gi[b3 + h], iz = gi[b3 + HH + h], inn = gi[b3 + 2 * HH + h];
  float hr = gh[b3 + h], hz = gh[b3 + HH + h], hn = gh[b3 + 2 * HH + h];
  float rr = sigmf(ir + hr);
  float z = sigmf(iz + hz);
  float nw = tanhf(inn + rr * hn);
  float o = (1.f - z) * nw + z * hprev[(size_t)r * HH + h];
  xo[(size_t)r * HH + h] = fmaxf(o, 0.f);
}

// head: lin2 over concat([out, t, p]) -> relu -> [B,64]
__global__ void lin2_k(const float* __restrict__ o, const float* __restrict__ t,
                       const float* __restrict__ p, const float* __restrict__ w,
                       const float* __restrict__ b, float* __restrict__ h64) {
  int i = blockIdx.x * 256 + threadIdx.x;
  if (i >= BBG * 64) return;
  int bi = i >> 6, oo = i & 63;
  const float* wr = w + oo * (HH + 2);
  const float* xr = o + (size_t)bi * HH;
  float s = b[oo];
  for (int k = 0; k < HH; ++k) s += xr[k] * wr[k];
  s += t[bi] * wr[HH] + p[bi] * wr[HH + 1];
  h64[i] = fmaxf(s, 0.f);
}

__global__ void lin3_k(const float* __restrict__ h64, const float* __restrict__ w,
                       const float* __restrict__ b, float* __restrict__ h32) {
  int i = blockIdx.x * 256 + threadIdx.x;
  if (i >= BBG * 32) return;
  int bi = i >> 5, oo = i & 31;
  const float* wr = w + oo * 64;
  const float* xr = h64 + (size_t)bi * 64;
  float s = b[oo];
  for (int k = 0; k < 64; ++k) s += xr[k] * wr[k];
  h32[i] = fmaxf(s, 0.f);
}

__global__ void lin4_k(const float* __restrict__ h32, const float* __restrict__ w,
                       const float* __restrict__ b, float* __restrict__ outp) {
  int bi = blockIdx.x * 256 + threadIdx.x;
  if (bi >= BBG) return;
  const float* xr = h32 + (size_t)bi * 32;
  float s = b[0];
  for (int k = 0; k < 32; ++k) s += xr[k] * w[k];
  outp[bi] = s;
}

// ---------------------------------------------------------------------------
static inline void launch_gemm(int act, const float* A, int lda, const float* W, int ldw,
                               const float* bias, float* C, int ldc, int M, int K, int Out,
                               hipStream_t st) {
  dim3 g((M / 16 + 3) / 4, Out / 64), b(128);
  switch (act) {
    case ACT_NONE:  gemm_k<ACT_NONE><<<g, b, 0, st>>>(A, lda, W, ldw, bias, C, ldc, M, K, Out); break;
    case ACT_LEAKY: gemm_k<ACT_LEAKY><<<g, b, 0, st>>>(A, lda, W, ldw, bias, C, ldc, M, K, Out); break;
    case ACT_ELU:   gemm_k<ACT_ELU><<<g, b, 0, st>>>(A, lda, W, ldw, bias, C, ldc, M, K, Out); break;
    default:        gemm_k<ACT_RELU><<<g, b, 0, st>>>(A, lda, W, ldw, bias, C, ldc, M, K, Out); break;
  }
}

extern "C" void kernel_launch(void* const* d_in, const int* in_sizes, int n_in,
                              void* d_out, int out_size, void* d_ws, size_t ws_size,
                              hipStream_t stream) {
  (void)in_sizes; (void)n_in; (void)out_size; (void)ws_size;
  const float* x        = (const float*)d_in[0];
  const int*   ei       = (const int*)d_in[1];
  const float* edge_attr= (const float*)d_in[2];
  const int*   batch    = (const int*)d_in[3];
  const float* t_in     = (const float*)d_in[4];
  const float* p_in     = (const float*)d_in[5];
  const float* lin1_w   = (const float*)d_in[6];
  const float* lin1_b   = (const float*)d_in[7];
  const float* g_att_l  = (const float*)d_in[8];
  const float* g_att_r  = (const float*)d_in[9];
  const float* g_lin1_w = (const float*)d_in[10];
  const float* g_lin2_w = (const float*)d_in[11];
  const float* g_bias   = (const float*)d_in[12];
  const float* gru0_wih = (const float*)d_in[13];
  const float* gru0_whh = (const float*)d_in[14];
  const float* gru0_bih = (const float*)d_in[15];
  const float* gru0_bhh = (const float*)d_in[16];
  const float* atom_w   = (const float*)d_in[17];
  const float* atom_as  = (const float*)d_in[18];
  const float* atom_ad  = (const float*)d_in[19];
  const float* atom_b   = (const float*)d_in[20];
  const float* agru_wih = (const float*)d_in[21];
  const float* agru_whh = (const float*)d_in[22];
  const float* agru_bih = (const float*)d_in[23];
  const float* agru_bhh = (const float*)d_in[24];
  const float* mol_w    = (const float*)d_in[25];
  const float* mol_as   = (const float*)d_in[26];
  const float* mol_ad   = (const float*)d_in[27];
  const float* mol_b    = (const float*)d_in[28];
  const float* mgru_wih = (const float*)d_in[29];
  const float* mgru_whh = (const float*)d_in[30];
  const float* mgru_bih = (const float*)d_in[31];
  const float* mgru_bhh = (const float*)d_in[32];
  const float* lin2_w   = (const float*)d_in[33];
  const float* lin2_b   = (const float*)d_in[34];
  const float* lin3_w   = (const float*)d_in[35];
  const float* lin3_b   = (const float*)d_in[36];
  const float* lin4_w   = (const float*)d_in[37];
  const float* lin4_b   = (const float*)d_in[38];
  float* outp = (float*)d_out;

  float* Wf = (float*)d_ws;
  size_t off = 0;
  auto alloc = [&](size_t n) { float* pp = Wf + off; off += n; return pp; };
  float* x0    = alloc((size_t)NN * HH);
  float* xw1   = alloc((size_t)NN * HH);
  float* helu  = alloc((size_t)NN * HH);
  float* acc   = alloc((size_t)NN * HH);
  float* xnA   = alloc((size_t)NN * HH);
  float* xnB   = alloc((size_t)NN * HH);
  float* gi    = alloc((size_t)NN * 3 * HH);
  float* gh    = alloc((size_t)NN * 3 * HH);
  float* araw  = alloc((size_t)EE);
  float* aexp  = alloc((size_t)EE);
  float* nmax  = alloc((size_t)NN);
  float* nsum  = alloc((size_t)NN);
  float* ssrc  = alloc((size_t)NN);
  float* sdstn = alloc((size_t)NN);
  float* wsv   = alloc(HH);
  float* wdv   = alloc(HH);
  float* msv   = alloc(HH);
  float* mdv   = alloc(HH);
  float* outA  = alloc((size_t)BBG * HH);
  float* outB  = alloc((size_t)BBG * HH);
  float* accg  = alloc((size_t)BBG * HH);
  float* heluB = alloc((size_t)BBG * HH);
  float* gmax  = alloc((size_t)BBG);
  float* gsum  = alloc((size_t)BBG);
  float* sdg   = alloc((size_t)BBG);
  float* h64   = alloc((size_t)BBG * 64);
  float* h32   = alloc((size_t)BBG * 32);

  auto fill = [&](float* pp, float v, size_t n) {
    fill_k<<<dim3((unsigned)((n + 255) / 256)), dim3(256), 0, stream>>>(pp, v, (int)n);
  };
  const float NEGBIG = -3.0e38f;
  const dim3 b256(256), b128(128);
  const dim3 gE256((EE + 255) / 256), gN256((NN * HH + 255) / 256);
  const dim3 gEw(EE / 4), gNw((NN + 3) / 4), gBw((BBG + 3) / 4);

  // ---- lin1: x0 = leaky(x @ lin1_w^T + lin1_b) ----
  launch_gemm(ACT_LEAKY, x, IND, lin1_w, IND, lin1_b, x0, HH, NN, IND, HH, stream);

  // ---- GATEConv ----
  launch_gemm(ACT_NONE, x0, HH, g_lin1_w, HH + EDD, nullptr, xw1, HH, NN, HH, HH, stream);
  dot2_k<<<gNw, b128, 0, stream>>>(x0, g_att_r, g_att_r, ssrc, sdstn, NN);  // ssrc = x0·att_r
  fill(nmax, NEGBIG, NN); fill(nsum, 0.f, NN); fill(acc, 0.f, (size_t)NN * HH);
  gate_edge1_k<<<gEw, b128, 0, stream>>>(ei, g_lin1_w, g_att_l, xw1, edge_attr, ssrc, araw, nmax);
  edge_exp_k<<<gE256, b256, 0, stream>>>(ei + EE, araw, nmax, aexp, nsum, EE);
  gate_edge3_k<<<gEw, b128, 0, stream>>>(ei, g_lin1_w, xw1, edge_attr, aexp, nsum, acc);
  launch_gemm(ACT_ELU, acc, HH, g_lin2_w, HH, g_bias, helu, HH, NN, HH, HH, stream);
  launch_gemm(ACT_NONE, helu, HH, gru0_wih, HH, gru0_bih, gi, 3 * HH, NN, HH, 3 * HH, stream);
  launch_gemm(ACT_NONE, x0, HH, gru0_whh, HH, gru0_bhh, gh, 3 * HH, NN, HH, 3 * HH, stream);
  gru_combine_k<<<gN256, b256, 0, stream>>>(gi, gh, x0, xnA, NN);

  // ---- atom GAT + GRU layers ----
  float* xn_cur = xnA; float* xn_nxt = xnB;
  for (int l = 0; l < LLAYERS; ++l) {
    const float* wl = atom_w + (size_t)l * HH * HH;
    matvecT_k<<<1, HH, 0, stream>>>(wl, atom_as + (size_t)l * HH, wsv);
    matvecT_k<<<1, HH, 0, stream>>>(wl, atom_ad + (size_t)l * HH, wdv);
    dot2_k<<<gNw, b128, 0, stream>>>(xn_cur, wsv, wdv, ssrc, sdstn, NN);
    fill(nmax, NEGBIG, NN); fill(nsum, 0.f, NN); fill(acc, 0.f, (size_t)NN * HH);
    edge_alpha_k<<<gE256, b256, 0, stream>>>(ei, ei + EE, ssrc, sdstn, araw, nmax, EE);
    edge_exp_k<<<gE256, b256, 0, stream>>>(ei + EE, araw, nmax, aexp, nsum, EE);
    edge_scatter_k<<<gEw, b128, 0, stream>>>(ei, ei + EE, aexp, nsum, xn_cur, acc, EE);
    launch_gemm(ACT_ELU, acc, HH, wl, HH, atom_b + (size_t)l * HH, helu, HH, NN, HH, HH, stream);
    launch_gemm(ACT_NONE, helu, HH, agru_wih + (size_t)l * 3 * HH * HH, HH,
                agru_bih + (size_t)l * 3 * HH, gi, 3 * HH, NN, HH, 3 * HH, stream);
    launch_gemm(ACT_NONE, xn_cur, HH, agru_whh + (size_t)l * 3 * HH * HH, HH,
                agru_bhh + (size_t)l * 3 * HH, gh, 3 * HH, NN, HH, 3 * HH, stream);
    gru_combine_k<<<gN256, b256, 0, stream>>>(gi, gh, xn_cur, xn_nxt, NN);
    float* tmp = xn_cur; xn_cur = xn_nxt; xn_nxt = tmp;
  }

  // ---- readout: out = relu(segment_sum(xn, batch)) ----
  fill(accg, 0.f, (size_t)BBG * HH);
  segsum_k<<<gNw, b128, 0, stream>>>(batch, xn_cur, accg, NN);
  relu_copy_k<<<dim3((BBG * HH + 255) / 256), b256, 0, stream>>>(accg, outA, BBG * HH);

  // mol attention node-side scalars (constant across timesteps)
  matvecT_k<<<1, HH, 0, stream>>>(mol_w, mol_as, msv);
  matvecT_k<<<1, HH, 0, stream>>>(mol_w, mol_ad, mdv);
  dot2_k<<<gNw, b128, 0, stream>>>(xn_cur, msv, msv, ssrc, sdstn, NN);  // ssrc = xn·(W^T a_s)

  float* out_cur = outA; float* out_nxt = outB;
  const dim3 gNn256((NN + 255) / 256);
  for (int ts = 0; ts < NTS; ++ts) {
    dot2_k<<<gBw, b128, 0, stream>>>(out_cur, mdv, mdv, sdg, gmax, BBG);  // sdg = out·(W^T a_d)
    fill(gmax, NEGBIG, BBG); fill(gsum, 0.f, BBG); fill(accg, 0.f, (size_t)BBG * HH);
    edge_alpha_k<<<gNn256, b256, 0, stream>>>(nullptr, batch, ssrc, sdg, araw, gmax, NN);
    edge_exp_k<<<gNn256, b256, 0, stream>>>(batch, araw, gmax, aexp, gsum, NN);
    edge_scatter_k<<<gNw, b128, 0, stream>>>(nullptr, batch, aexp, gsum, xn_cur, accg, NN);
    launch_gemm(ACT_ELU, accg, HH, mol_w, HH, mol_b, heluB, HH, BBG, HH, HH, stream);
    launch_gemm(ACT_NONE, heluB, HH, mgru_wih, HH, mgru_bih, gi, 3 * HH, BBG, HH, 3 * HH, stream);
    launch_gemm(ACT_NONE, out_cur, HH, mgru_whh, HH, mgru_bhh, gh, 3 * HH, BBG, HH, 3 * HH, stream);
    gru_combine_k<<<dim3((BBG * HH + 255) / 256), b256, 0, stream>>>(gi, gh, out_cur, out_nxt, BBG);
    float* tmp = out_cur; out_cur = out_nxt; out_nxt = tmp;
  }

  // ---- MLP head ----
  lin2_k<<<dim3((BBG * 64 + 255) / 256), b256, 0, stream>>>(out_cur, t_in, p_in, lin2_w, lin2_b, h64);
  lin3_k<<<dim3((BBG * 32 + 255) / 256), b256, 0, stream>>>(h64, lin3_w, lin3_b, h32);
  lin4_k<<<dim3((BBG + 255) / 256), b256, 0, stream>>>(h32, lin4_w, lin4_b, outp);
}